// mLSTMCell_39436389712570
// MI455X (gfx1250) — compile-verified
//
#include <hip/hip_runtime.h>
#include <hip/hip_bf16.h>
#include <math.h>

// ---------------------------------------------------------------------------
// mLSTM single-step cell for MI455X (gfx1250).
//   B=64, NH=8, HD=256, H=2048.
// Bandwidth-bound: c (134MB) read + c_new (134MB) written dominate (~11.5us
// at 23.3 TB/s). Gates GEMM (64 x 16 x 6144) uses V_WMMA_F32_16X16X4_F32
// with a branchless, pipelineable K loop.
// ---------------------------------------------------------------------------

typedef __attribute__((ext_vector_type(2))) float v2f;
typedef __attribute__((ext_vector_type(4))) float v4f;
typedef __attribute__((ext_vector_type(8))) float v8f;

#define B_   64
#define NH_  8
#define HD_  256
#define H_   2048

// ---------------------------------------------------------------------------
// Kernel 1: gates via WMMA f32 16x16x4.
//   24 waves = 4 M-tiles (16 rows of qkv each) x 6 K-chunks of 1024.
//   Each K-chunk lies entirely inside one of q/k/v -> wave-uniform source
//   pointer, no divergent branches in the inner loop. The wi|wf column
//   select is a loop-invariant per-lane base pointer.
//   Partial 16x16 D tiles merged with LDS float atomics, then the
//   log-sigmoid / stabilizer epilogue writes m_new and (f', i', exp(-m_new)).
// ---------------------------------------------------------------------------
__global__ __launch_bounds__(768) void gates_kernel(
    const float* __restrict__ q, const float* __restrict__ k,
    const float* __restrict__ v, const float* __restrict__ m,
    const float* __restrict__ wi, const float* __restrict__ wib,
    const float* __restrict__ wf, const float* __restrict__ wfb,
    float* __restrict__ m_new_out, float* __restrict__ ws) {
  __shared__ float g[4 * 256];   // 4 M-tiles of 16x16 accumulators

  const int tid = threadIdx.x;
  for (int i = tid; i < 4 * 256; i += 768) g[i] = 0.0f;
  __syncthreads();

  const int wid  = tid >> 5;
  const int lane = tid & 31;
  const int mt   = wid & 3;        // M tile: rows 16*mt .. 16*mt+15
  const int kc   = wid >> 2;       // K chunk: 1024 wide, 0..5
  const int src  = kc >> 1;        // 0=q, 1=k, 2=v (wave-uniform)
  const int soff = (kc & 1) * 1024;
  const int row  = mt * 16 + (lane & 15);
  const int ncol = lane & 15;
  const int half = lane >> 4;      // 0: K={0,1}, 1: K={2,3} within a step

  // Wave-uniform source pointer (q/k/v), loop-invariant.
  const float* __restrict__ srcp = (src == 0) ? q : (src == 1) ? k : v;
  // Per-lane A base: row of qkv, offset into this K-chunk, K-pair select.
  const float* __restrict__ aptr = srcp + row * H_ + soff + 2 * half;
  // Per-lane B base: column of [wi | wf] (6144x16), row stride 8 floats.
  const int wbase = src * H_ + soff + 2 * half;      // global K row
  const float* __restrict__ bptr =
      ((ncol < NH_) ? (wi + ncol) : (wf + (ncol - NH_))) + (size_t)wbase * NH_;

  v8f acc = {};
#pragma unroll 4
  for (int kk = 0; kk < 1024; kk += 4) {
    v2f a, bfrag;
    a.x     = aptr[kk];
    a.y     = aptr[kk + 1];
    bfrag.x = bptr[(size_t)kk * NH_];
    bfrag.y = bptr[(size_t)(kk + 1) * NH_];
    acc = __builtin_amdgcn_wmma_f32_16x16x4_f32(
        /*neg_a=*/false, a, /*neg_b=*/false, bfrag,
        /*c_mod=*/(short)0, acc, /*reuse_a=*/false, /*reuse_b=*/false);
  }

  // D 16x16 layout: VGPR r -> M = r + 8*half, lane%16 -> N
#pragma unroll
  for (int r = 0; r < 8; ++r) {
    atomicAdd(&g[mt * 256 + (r + 8 * half) * 16 + ncol], acc[r]);
  }
  __syncthreads();

  if (tid < B_ * NH_) {                       // 512 scalar epilogues
    const int b   = tid >> 3;
    const int h   = tid & 7;
    const int mt2 = b >> 4, r2 = b & 15;
    const float gi = g[mt2 * 256 + r2 * 16 + h];
    const float gf = g[mt2 * 256 + r2 * 16 + 8 + h];
    const float it = gi + wib[h];
    const float ft = gf + wfb[h];
    // log sigmoid(ft) = -softplus(-ft), numerically stable
    const float sp    = log1pf(expf(-fabsf(ft))) + fmaxf(-ft, 0.0f);
    const float logf_ = -sp;
    const int   ph    = b * NH_ + h;
    const float m_old = m[ph];
    const float mn = fmaxf(logf_ + m_old, it);
    m_new_out[ph] = mn;
    ws[ph * 3 + 0] = expf(logf_ + m_old - mn);  // f'
    ws[ph * 3 + 1] = expf(it - mn);             // i'
    ws[ph * 3 + 2] = expf(-mn);                 // denominator floor
  }
}

// ---------------------------------------------------------------------------
// Kernel 2: fused streaming update. One block per (b, h) 256x256 tile.
//   n_new + q.n_new reduction, then non-temporal float4 stream of
//   c -> c_new with fused nominator accumulation, then per-head layernorm.
// ---------------------------------------------------------------------------
__global__ __launch_bounds__(256) void main_kernel(
    const float* __restrict__ q, const float* __restrict__ k,
    const float* __restrict__ v, const float* __restrict__ c,
    const float* __restrict__ n_in, const float* __restrict__ ln_scale,
    const float* __restrict__ ws,
    float* __restrict__ out_c, float* __restrict__ out_n,
    float* __restrict__ out_h) {
  __shared__ float lq[HD_], lk[HD_], lv[HD_];
  __shared__ float red[1024];

  const int t  = threadIdx.x;
  const int bh = blockIdx.x;           // 0..511
  const int b  = bh >> 3;
  const int h  = bh & 7;
  const int base = b * H_ + h * HD_;

  const float qv = q[base + t];
  const float kv = k[base + t] * 0.0625f;   // 1/sqrt(256)
  lq[t] = qv;
  lk[t] = kv;
  lv[t] = v[base + t];

  const float fpr  = ws[bh * 3 + 0];
  const float ipr  = ws[bh * 3 + 1];
  const float denf = ws[bh * 3 + 2];

  // n_new and q . n_new
  const float nn = fpr * n_in[bh * HD_ + t] + ipr * kv;
  out_n[bh * HD_ + t] = nn;
  red[t] = qv * nn;
  __syncthreads();
  for (int s = 128; s > 0; s >>= 1) {
    if (t < s) red[t] += red[t + s];
    __syncthreads();
  }
  const float den     = fmaxf(fabsf(red[0]), denf);
  const float inv_den = 1.0f / (den + 1e-6f);
  __syncthreads();

  // Stream the 256x256 tile: 64 column groups (v4f) x 4 row chunks.
  // Touch-once data -> non-temporal loads/stores (don't thrash 192MB L2).
  const int cg = t & 63;
  const int rc = t >> 6;
  const v4f* __restrict__ cin  = (const v4f*)c + (size_t)bh * 16384;
  v4f* __restrict__       cout = (v4f*)out_c   + (size_t)bh * 16384;

  v4f lvv;
  lvv.x = lv[cg * 4 + 0];
  lvv.y = lv[cg * 4 + 1];
  lvv.z = lv[cg * 4 + 2];
  lvv.w = lv[cg * 4 + 3];
  v4f acc = {};
  const int i0 = rc * 64;
#pragma unroll 4
  for (int i = i0; i < i0 + 64; ++i) {
    const v4f   cv = __builtin_nontemporal_load(&cin[i * 64 + cg]);
    const float s1 = ipr * lk[i];
    v4f cn;
    cn.x = fpr * cv.x + s1 * lvv.x;
    cn.y = fpr * cv.y + s1 * lvv.y;
    cn.z = fpr * cv.z + s1 * lvv.z;
    cn.w = fpr * cv.w + s1 * lvv.w;
    __builtin_nontemporal_store(cn, &cout[i * 64 + cg]);
    const float qi = lq[i];
    acc.x += qi * cn.x;
    acc.y += qi * cn.y;
    acc.z += qi * cn.z;
    acc.w += qi * cn.w;
  }

  // Merge the 4 row-chunk partials per column.
  red[rc * 256 + cg * 4 + 0] = acc.x;
  red[rc * 256 + cg * 4 + 1] = acc.y;
  red[rc * 256 + cg * 4 + 2] = acc.z;
  red[rc * 256 + cg * 4 + 3] = acc.w;
  __syncthreads();
  const float nom = red[t] + red[256 + t] + red[512 + t] + red[768 + t];
  const float ht  = nom * inv_den;
  __syncthreads();

  // Layernorm over HD (population variance), per-head scale, no bias.
  red[t] = ht;
  __syncthreads();
  for (int s = 128; s > 0; s >>= 1) {
    if (t < s) red[t] += red[t + s];
    __syncthreads();
  }
  const float mu = red[0] * (1.0f / 256.0f);
  __syncthreads();
  const float d = ht - mu;
  red[t] = d * d;
  __syncthreads();
  for (int s = 128; s > 0; s >>= 1) {
    if (t < s) red[t] += red[t + s];
    __syncthreads();
  }
  const float var = red[0] * (1.0f / 256.0f);
  const float hn  = d * rsqrtf(var + 1e-6f) * ln_scale[h * HD_ + t];
  out_h[b * H_ + h * HD_ + t] = hn;
}

// ---------------------------------------------------------------------------
extern "C" void kernel_launch(void* const* d_in, const int* in_sizes, int n_in,
                              void* d_out, int out_size, void* d_ws, size_t ws_size,
                              hipStream_t stream) {
  const float* q   = (const float*)d_in[0];
  const float* k   = (const float*)d_in[1];
  const float* v   = (const float*)d_in[2];
  const float* c   = (const float*)d_in[3];
  const float* n   = (const float*)d_in[4];
  const float* m   = (const float*)d_in[5];
  const float* wik = (const float*)d_in[6];
  const float* wib = (const float*)d_in[7];
  const float* wfk = (const float*)d_in[8];
  const float* wfb = (const float*)d_in[9];
  const float* lns = (const float*)d_in[10];

  float* out   = (float*)d_out;
  float* out_c = out;                                    // 64*8*256*256
  float* out_n = out_c + (size_t)B_ * NH_ * HD_ * HD_;   // 64*8*256
  float* out_m = out_n + (size_t)B_ * NH_ * HD_;         // 64*8
  float* out_h = out_m + (size_t)B_ * NH_;               // 64*2048

  float* ws = (float*)d_ws;                              // 512*3 floats

  gates_kernel<<<1, 768, 0, stream>>>(q, k, v, m, wik, wib, wfk, wfb,
                                      out_m, ws);
  main_kernel<<<B_ * NH_, HD_, 0, stream>>>(q, k, v, c, n, lns, ws,
                                            out_c, out_n, out_h);
}